// Eagle3Attention_12360915878787
// MI455X (gfx1250) — compile-verified
//
#include <hip/hip_runtime.h>

typedef __attribute__((ext_vector_type(2))) float v2f;
typedef __attribute__((ext_vector_type(8))) float v8f;
typedef __attribute__((ext_vector_type(4))) int v4i;

constexpr int B_ = 4, S_ = 64, H_ = 4096, NQ_ = 32, NKV_ = 8, D_ = 128, MAXLEN_ = 4096;
constexpr int G_ = NQ_ / NKV_;          // 4
constexpr int IN_DIM = 2 * H_;          // 8192
constexpr int ROWS = B_ * S_;           // 256
constexpr float SCALE_ = 0.08838834764831843f;  // 1/sqrt(128)

// ---------------------------------------------------------------------------
// 16-byte global -> LDS copy. Uses CDNA5 async-to-LDS (ASYNCcnt) when the
// toolchain exposes the builtin; falls back to a synchronous float4 copy.
// Builtin signature (from clang diagnostic): (int4 AS1*, int4 AS3*, Ii, Ii).
// ---------------------------------------------------------------------------
#if defined(__AMDGCN__) && __has_builtin(__builtin_amdgcn_global_load_async_to_lds_b128)
#define ASYNC_LDS 1
#else
#define ASYNC_LDS 0
#endif

typedef __attribute__((address_space(1))) v4i* gptr4;
typedef __attribute__((address_space(3))) v4i* lptr4;

__device__ __forceinline__ void cp16(float* dst_lds, const float* src_global) {
#if ASYNC_LDS
  __builtin_amdgcn_global_load_async_to_lds_b128(
      (gptr4)const_cast<float*>(src_global), (lptr4)dst_lds, 0, 0);
#else
  *(float4*)dst_lds = *(const float4*)src_global;
#endif
}

__device__ __forceinline__ void wait_cp() {
#if ASYNC_LDS
#if __has_builtin(__builtin_amdgcn_s_wait_asynccnt)
  __builtin_amdgcn_s_wait_asynccnt(0);
#else
  asm volatile("s_wait_asynccnt 0" ::: "memory");
#endif
#endif
}

// ---------------------------------------------------------------------------
// Tiled FP32 WMMA GEMM: C(MxN) = A(MxK) @ B(KxN), row-major.
// Block: 256 threads = 8 waves, block tile 128(M) x 64(N); wave tile 16x64.
// B k-panels (32x64) staged in LDS (stride 72 -> conflict-free frag reads),
// double-buffered via async-to-LDS.  M%128==0, N%64==0, K%32==0 assumed.
// A-frag lane map (V_WMMA_F32_16X16X4_F32): m = lane&15, k = 2*(lane>>4)+v.
// B-frag: n = lane&15, k = 2*(lane>>4)+v.  C-frag: m = v + 8*(lane>>4), n = lane&15.
// ---------------------------------------------------------------------------
constexpr int KCH = 32;    // k-chunk staged per buffer
constexpr int BSTR = 72;   // LDS row stride (floats) for B panel

__global__ __launch_bounds__(256)
void wmma_gemm_f32(const float* __restrict__ A, const float* __restrict__ Bm,
                   float* __restrict__ C, int M, int N, int K) {
  __shared__ float Bs[2][KCH * BSTR];

  const int t = threadIdx.x;
  const int lane = t & 31;
  const int wave = t >> 5;
  const int h = lane >> 4;
  const int l16 = lane & 15;
  const int m0 = blockIdx.y * 128 + wave * 16;
  const int n0 = blockIdx.x * 64;

  // stage B chunk 0: 32x64 floats = 512 float4, 2 per thread
  {
    const float* src = Bm + n0;
#pragma unroll
    for (int j = 0; j < 2; ++j) {
      int idx4 = t + j * 256;
      int row = idx4 >> 4, col = (idx4 & 15) * 4;
      cp16(&Bs[0][row * BSTR + col], src + (size_t)row * N + col);
    }
  }

  v8f acc0 = {}, acc1 = {}, acc2 = {}, acc3 = {};
  const float* Arow = A + (size_t)(m0 + l16) * K + 2 * h;
  const int nchunks = K / KCH;

  for (int c = 0; c < nchunks; ++c) {
    const int cur = c & 1;
    wait_cp();
    __syncthreads();   // tile c visible to all; all reads of buffer cur^1 retired

    if (c + 1 < nchunks) {
      const float* src = Bm + (size_t)(c + 1) * KCH * N + n0;
#pragma unroll
      for (int j = 0; j < 2; ++j) {
        int idx4 = t + j * 256;
        int row = idx4 >> 4, col = (idx4 & 15) * 4;
        cp16(&Bs[cur ^ 1][row * BSTR + col], src + (size_t)row * N + col);
      }
    }

    const float* Ap = Arow + c * KCH;
    const float* bs = Bs[cur];
    __builtin_prefetch(Ap + 8 * KCH, 0, 1);
#pragma unroll
    for (int kk = 0; kk < KCH; kk += 4) {
      v2f a = *(const v2f*)(Ap + kk);
      const float* bp = bs + (kk + 2 * h) * BSTR + l16;
      v2f b0, b1, b2, b3;
      b0.x = bp[0];  b0.y = bp[BSTR];
      b1.x = bp[16]; b1.y = bp[BSTR + 16];
      b2.x = bp[32]; b2.y = bp[BSTR + 32];
      b3.x = bp[48]; b3.y = bp[BSTR + 48];
      acc0 = __builtin_amdgcn_wmma_f32_16x16x4_f32(false, a, false, b0, (short)0, acc0, false, false);
      acc1 = __builtin_amdgcn_wmma_f32_16x16x4_f32(false, a, false, b1, (short)0, acc1, false, false);
      acc2 = __builtin_amdgcn_wmma_f32_16x16x4_f32(false, a, false, b2, (short)0, acc2, false, false);
      acc3 = __builtin_amdgcn_wmma_f32_16x16x4_f32(false, a, false, b3, (short)0, acc3, false, false);
    }
  }

  float* Ccol = C + n0 + l16;
#pragma unroll
  for (int v = 0; v < 8; ++v) {
    size_t r = (size_t)(m0 + v + 8 * h) * N;
    Ccol[r + 0]  = acc0[v];
    Ccol[r + 16] = acc1[v];
    Ccol[r + 32] = acc2[v];
    Ccol[r + 48] = acc3[v];
  }
}

// ---------------------------------------------------------------------------
// Bulk float4 copy: past_key_value -> present_key_value output region.
// ---------------------------------------------------------------------------
__global__ void copy_f4(const float4* __restrict__ src, float4* __restrict__ dst, int n4) {
  int i = blockIdx.x * blockDim.x + threadIdx.x;
  if (i < n4) dst[i] = src[i];
}

// ---------------------------------------------------------------------------
// RoPE on Q in place.  q layout: (B,S,NQ,D).  rope table: (MAXLEN, 2, D/2).
// ---------------------------------------------------------------------------
__global__ void rope_q(float* __restrict__ q, const float* __restrict__ rope,
                       const int* __restrict__ ctxlen) {
  int i = blockIdx.x * blockDim.x + threadIdx.x;   // B*S*NQ*64 threads
  int dh = i & 63;
  int t = i >> 6;
  int hq = t & 31; t >>= 5;
  int s = t & 63;  t >>= 6;
  int b = t;
  int pos = ctxlen[b] + s;
  float c = rope[(size_t)pos * D_ + dh];
  float sn = rope[(size_t)pos * D_ + 64 + dh];
  size_t base = ((size_t)(b * S_ + s) * NQ_ + hq) * D_;
  float x1 = q[base + dh], x2 = q[base + 64 + dh];
  q[base + dh]      = x1 * c - x2 * sn;
  q[base + 64 + dh] = x2 * c + x1 * sn;
}

// ---------------------------------------------------------------------------
// RoPE on K + scatter K,V into present cache (B,2,NKV,MAXLEN,D) at pos=ctx+s.
// ---------------------------------------------------------------------------
__global__ void rope_scatter_kv(const float* __restrict__ kraw, const float* __restrict__ vraw,
                                const float* __restrict__ rope, const int* __restrict__ ctxlen,
                                float* __restrict__ cache) {
  int i = blockIdx.x * blockDim.x + threadIdx.x;   // B*S*NKV*64 threads
  int dh = i & 63;
  int t = i >> 6;
  int hk = t & 7; t >>= 3;
  int s = t & 63; t >>= 6;
  int b = t;
  int pos = ctxlen[b] + s;
  float c = rope[(size_t)pos * D_ + dh];
  float sn = rope[(size_t)pos * D_ + 64 + dh];
  size_t ibase = ((size_t)(b * S_ + s) * NKV_ + hk) * D_;
  float x1 = kraw[ibase + dh], x2 = kraw[ibase + 64 + dh];
  size_t kb = (((size_t)(b * 2 + 0) * NKV_ + hk) * MAXLEN_ + pos) * (size_t)D_;
  cache[kb + dh]      = x1 * c - x2 * sn;
  cache[kb + 64 + dh] = x2 * c + x1 * sn;
  size_t vb = (((size_t)(b * 2 + 1) * NKV_ + hk) * MAXLEN_ + pos) * (size_t)D_;
  cache[vb + dh]      = vraw[ibase + dh];
  cache[vb + 64 + dh] = vraw[ibase + 64 + dh];
}

// ---------------------------------------------------------------------------
// Flash attention over full MAXLEN cache.
// Grid: B*NKV blocks; block = 512 threads = 16 waves = (G=4) x (4 q-tiles of 16).
// K/V 16x128 tiles double-buffered in LDS via async-to-LDS; padded stride 132.
// Scores computed transposed (S^T = K_tile * Q^T) so softmax over p is a
// per-lane reduction + one shfl_xor(16).  P restaged via LDS into A-frag form.
// One __syncthreads per tile: it both publishes the arrived tile and retires
// all reads of the buffer that the next async load will overwrite.
// ---------------------------------------------------------------------------
__global__ __launch_bounds__(512)
void flash_attn(const float* __restrict__ q,       // (B,S,NQ,D) rope'd
                const float* __restrict__ cache,   // (B,2,NKV,MAXLEN,D)
                const int* __restrict__ ctxlen,
                float* __restrict__ ctx) {         // (B,S,NQ,D)
  constexpr int KSTR = 132;
  constexpr int NT = MAXLEN_ / 16;
  __shared__ float Kt[2][16 * KSTR];
  __shared__ float Vt[2][16 * KSTR];
  __shared__ float Pst[16 * 256];   // per-wave 16x16 P^T staging
  __shared__ float Ast[16 * 16];    // per-wave alpha / l broadcast

  const int b = blockIdx.x >> 3;
  const int kvh = blockIdx.x & 7;
  const int t = threadIdx.x;
  const int lane = t & 31;
  const int wave = t >> 5;
  const int h = lane >> 4;
  const int l16 = lane & 15;
  const int g = wave >> 2;
  const int qt = wave & 3;
  const int qrow = qt * 16 + l16;
  const int CL = ctxlen[b];

  const float* kbaseg = cache + (((size_t)(b * 2 + 0) * NKV_ + kvh) * MAXLEN_) * (size_t)D_;
  const float* vbaseg = cache + (((size_t)(b * 2 + 1) * NKV_ + kvh) * MAXLEN_) * (size_t)D_;

  const int lrow16 = t >> 5;          // 16B-chunk row in tile
  const int lcol = (t & 31) * 4;      // 16B-chunk col

  // stage tile 0
  cp16(&Kt[0][lrow16 * KSTR + lcol], kbaseg + lrow16 * D_ + lcol);
  cp16(&Vt[0][lrow16 * KSTR + lcol], vbaseg + lrow16 * D_ + lcol);

  // Q held in registers: lane owns row qrow, d in {4j+2h, 4j+2h+1}.
  v2f qreg[32];
  {
    const float* qp = q + ((size_t)(b * S_ + qrow) * NQ_ + (kvh * G_ + g)) * D_ + 2 * h;
#pragma unroll
    for (int j = 0; j < 32; ++j) qreg[j] = *(const v2f*)(qp + 4 * j);
  }

  v8f o[8] = {};
  float mrow = -1.0e9f, lrow = 0.0f;

  for (int it = 0; it < NT; ++it) {
    const int cur = it & 1;
    wait_cp();
    __syncthreads();

    if (it + 1 < NT) {
      const size_t goff = (size_t)(it + 1) * 16 * D_ + lrow16 * D_ + lcol;
      cp16(&Kt[cur ^ 1][lrow16 * KSTR + lcol], kbaseg + goff);
      cp16(&Vt[cur ^ 1][lrow16 * KSTR + lcol], vbaseg + goff);
    }

    // --- S^T(p,q) = sum_d K(p,d) * Q(q,d) ---
    v8f sfr = {};
#pragma unroll
    for (int kk = 0; kk < 32; ++kk) {
      v2f a = *(const v2f*)&Kt[cur][l16 * KSTR + 4 * kk + 2 * h];
      sfr = __builtin_amdgcn_wmma_f32_16x16x4_f32(false, a, false, qreg[kk], (short)0, sfr,
                                                  false, false);
    }

    // --- scale + causal/context mask (analytic tril) ---
    float tmax = -1.0e30f;
#pragma unroll
    for (int v = 0; v < 8; ++v) {
      int pglob = it * 16 + v + 8 * h;
      float sv = sfr[v] * SCALE_;
      bool valid = (pglob < CL) || ((pglob < CL + S_) && ((pglob - CL) <= qrow));
      sv = valid ? sv : -1.0e9f;
      sfr[v] = sv;
      tmax = fmaxf(tmax, sv);
    }
    tmax = fmaxf(tmax, __shfl_xor(tmax, 16, 32));

    float mnew = fmaxf(mrow, tmax);
    float alpha = __expf(mrow - mnew);
    mrow = mnew;

    float tsum = 0.0f;
#pragma unroll
    for (int v = 0; v < 8; ++v) {
      float e = __expf(sfr[v] - mnew);
      sfr[v] = e;
      tsum += e;
    }
    tsum += __shfl_xor(tsum, 16, 32);
    lrow = lrow * alpha + tsum;

    // --- stage alpha (per q) and P^T for layout conversion (per-wave region) ---
    if (h == 0) Ast[wave * 16 + l16] = alpha;
#pragma unroll
    for (int v = 0; v < 8; ++v) Pst[wave * 256 + (v + 8 * h) * 16 + l16] = sfr[v];

    // --- rescale O by alpha[row] (LDS broadcast) ---
    float av[8];
#pragma unroll
    for (int v = 0; v < 8; ++v) av[v] = Ast[wave * 16 + v + 8 * h];
#pragma unroll
    for (int f = 0; f < 8; ++f)
#pragma unroll
      for (int v = 0; v < 8; ++v) o[f][v] *= av[v];

    // --- O(q,d) += P(q,p) * V(p,d) ---
#pragma unroll
    for (int pk = 0; pk < 4; ++pk) {
      v2f a;
      a.x = Pst[wave * 256 + (pk * 4 + 2 * h) * 16 + l16];
      a.y = Pst[wave * 256 + (pk * 4 + 2 * h + 1) * 16 + l16];
#pragma unroll
      for (int f = 0; f < 8; ++f) {
        v2f bb;
        bb.x = Vt[cur][(pk * 4 + 2 * h) * KSTR + f * 16 + l16];
        bb.y = Vt[cur][(pk * 4 + 2 * h + 1) * KSTR + f * 16 + l16];
        o[f] = __builtin_amdgcn_wmma_f32_16x16x4_f32(false, a, false, bb, (short)0, o[f],
                                                     false, false);
      }
    }
  }

  // --- normalize by l and write ctx (B,S,NQ,D) ---
  if (h == 0) Ast[wave * 16 + l16] = lrow;
  float inv[8];
#pragma unroll
  for (int v = 0; v < 8; ++v) inv[v] = 1.0f / Ast[wave * 16 + v + 8 * h];

#pragma unroll
  for (int f = 0; f < 8; ++f)
#pragma unroll
    for (int v = 0; v < 8; ++v) {
      size_t off = ((size_t)(b * S_ + qt * 16 + v + 8 * h) * NQ_ + (kvh * G_ + g)) * D_ +
                   f * 16 + l16;
      ctx[off] = o[f][v] * inv[v];
    }
}

// ---------------------------------------------------------------------------
extern "C" void kernel_launch(void* const* d_in, const int* in_sizes, int n_in,
                              void* d_out, int out_size, void* d_ws, size_t ws_size,
                              hipStream_t stream) {
  (void)in_sizes; (void)n_in; (void)out_size; (void)ws_size;
  const float* hidden = (const float*)d_in[0];
  const float* past   = (const float*)d_in[1];
  const float* rope   = (const float*)d_in[2];
  const int*   ctxlen = (const int*)d_in[3];
  const float* Wq = (const float*)d_in[7];
  const float* Wk = (const float*)d_in[8];
  const float* Wv = (const float*)d_in[9];
  const float* Wo = (const float*)d_in[10];

  float* out = (float*)d_out;
  float* cacheOut = out + (size_t)ROWS * H_;            // present_key_value region

  float* q_ws   = (float*)d_ws;                          // 256*4096
  float* k_ws   = q_ws + (size_t)ROWS * NQ_ * D_;        // 256*1024
  float* v_ws   = k_ws + (size_t)ROWS * NKV_ * D_;       // 256*1024
  float* ctx_ws = v_ws + (size_t)ROWS * NKV_ * D_;       // 256*4096

  dim3 blk(256);

  // 1) QKV projections (fp32 WMMA GEMM)
  wmma_gemm_f32<<<dim3((NQ_ * D_) / 64, ROWS / 128), blk, 0, stream>>>(
      hidden, Wq, q_ws, ROWS, NQ_ * D_, IN_DIM);
  wmma_gemm_f32<<<dim3((NKV_ * D_) / 64, ROWS / 128), blk, 0, stream>>>(
      hidden, Wk, k_ws, ROWS, NKV_ * D_, IN_DIM);
  wmma_gemm_f32<<<dim3((NKV_ * D_) / 64, ROWS / 128), blk, 0, stream>>>(
      hidden, Wv, v_ws, ROWS, NKV_ * D_, IN_DIM);

  // 2) present_key_value = copy of past cache
  int n4 = (B_ * 2 * NKV_ * MAXLEN_ * D_) / 4;
  copy_f4<<<(n4 + 255) / 256, 256, 0, stream>>>((const float4*)past, (float4*)cacheOut, n4);

  // 3) RoPE q in place; RoPE k + scatter k/v into present cache
  rope_q<<<(ROWS * NQ_ * 64) / 256, 256, 0, stream>>>(q_ws, rope, ctxlen);
  rope_scatter_kv<<<(ROWS * NKV_ * 64) / 256, 256, 0, stream>>>(k_ws, v_ws, rope, ctxlen,
                                                                cacheOut);

  // 4) flash attention over full cache
  flash_attn<<<B_ * NKV_, 512, 0, stream>>>(q_ws, cacheOut, ctxlen, ctx_ws);

  // 5) output projection
  wmma_gemm_f32<<<dim3(H_ / 64, ROWS / 128), blk, 0, stream>>>(
      ctx_ws, Wo, out, ROWS, H_, NQ_ * D_);
}